// SEGNN_22110491640397
// MI455X (gfx1250) — compile-verified
//
#include <hip/hip_runtime.h>

// ---------------------------------------------------------------------------
// SEGNN forward for MI455X (gfx1250, wave32).
// Matrix path: V_WMMA_F32_16X16X4_F32 (fp32 A/B, fp32 accum) — preserves the
// reference's fp32 numerics while using the CDNA5 matrix pipe.
// One wave per workgroup processes 16 rows (edges or nodes) as the M dim.
// ---------------------------------------------------------------------------

typedef __attribute__((ext_vector_type(2))) float v2f;
typedef __attribute__((ext_vector_type(8))) float v8f;

struct TPW {
  const float* w00;   // [K0][N0]
  const float* w110;  // [K1][N0]
  const float* b0;    // [N0]
  const float* w01;   // [K0][N1]
  const float* w10;   // [K1][N1]
  const float* w11;   // [K1][N1]
};

// C += A(16 x K) * B(K x 16), K stepped by 4 with zero padding.
// A: LDS, row-major, stride lda.  B: global, row-major [K][ldb], column tile at col0.
// Fragment layouts per CDNA5 ISA 7.12.2:
//   A: lane m (=lane&15) holds A[m][2h], A[m][2h+1]   (h = lane>>4)
//   B: lane   holds B[2h+k][lane&15]
//   C: vgpr j holds row j+8h, col = lane&15
__device__ __forceinline__ v8f wmma_acc(const float* __restrict__ A, int lda, int K,
                                        const float* __restrict__ B, int ldb, int col0,
                                        v8f c) {
  const int m  = threadIdx.x & 15;
  const int h  = (threadIdx.x >> 4) & 1;
  for (int k = 0; k < K; k += 4) {
    int ka = k + 2 * h;
    v2f a; a.x = 0.f; a.y = 0.f;
    v2f b; b.x = 0.f; b.y = 0.f;
    if (ka < K)     { a.x = A[m * lda + ka];     b.x = B[(size_t)ka * ldb + col0 + m]; }
    if (ka + 1 < K) { a.y = A[m * lda + ka + 1]; b.y = B[(size_t)(ka + 1) * ldb + col0 + m]; }
    c = __builtin_amdgcn_wmma_f32_16x16x4_f32(false, a, false, b, (short)0, c, false, false);
  }
  return c;
}

// Steerable tensor product for 16 rows, one wave.
//  X0: LDS [16][ldx0] raw scalars (K0)      XV: LDS [3][16][ldxv] raw vectors (K1)
//  a0: 16 floats in LDS, a1: 16x3 floats in LDS
//  P : LDS scratch [16][ldp]  (ldp >= max(K0+K1, 2*K1))
//  T : LDS scratch [16][ldt]  (ldt >= N1)
//  Sout: LDS [16][lds_] (N0 cols)    Vout: LDS [3][16][ldvo] (N1 cols)
__device__ void tp_wave(const float* __restrict__ X0, int ldx0, int K0,
                        const float* __restrict__ XV, int ldxv, int K1,
                        const float* __restrict__ a0, const float* __restrict__ a1,
                        float* P, int ldp, float* T, int ldt,
                        float* Sout, int lds_, int N0,
                        float* Vout, int ldvo, int N1,
                        TPW w) {
  const int lane = threadIdx.x & 31;
  const int r = lane & 15;
  const int h = lane >> 4;

  // ---- scalar path panel: [ x0*a0 | dot(x1, a1) ]
  for (int k = h; k < K0; k += 2) P[r * ldp + k] = X0[r * ldx0 + k] * a0[r];
  for (int k = h; k < K1; k += 2) {
    float dsum = XV[(0 * 16 + r) * ldxv + k] * a1[r * 3 + 0]
               + XV[(1 * 16 + r) * ldxv + k] * a1[r * 3 + 1]
               + XV[(2 * 16 + r) * ldxv + k] * a1[r * 3 + 2];
    P[r * ldp + K0 + k] = dsum;
  }
  __syncthreads();

  // ---- s_out = panel @ [w00 ; w110] + b0
  for (int c0 = 0; c0 < N0; c0 += 16) {
    float bias = w.b0[c0 + r];
    v8f c;
    for (int j = 0; j < 8; ++j) c[j] = bias;
    c = wmma_acc(P,      ldp, K0, w.w00,  N0, c0, c);
    c = wmma_acc(P + K0, ldp, K1, w.w110, N0, c0, c);
    for (int j = 0; j < 8; ++j) Sout[(j + 8 * h) * lds_ + c0 + r] = c[j];
  }

  // ---- t = x0 @ w01
  for (int c0 = 0; c0 < N1; c0 += 16) {
    v8f c = {};
    c = wmma_acc(X0, ldx0, K0, w.w01, N1, c0, c);
    for (int j = 0; j < 8; ++j) T[(j + 8 * h) * ldt + c0 + r] = c[j];
  }

  // ---- vector path: v_d = t*a1_d + [x1_d*a0 | cross_d] @ [w10 ; w11]
  for (int d = 0; d < 3; ++d) {
    const int d1 = (d == 2) ? 0 : d + 1;   // (d+1)%3
    const int d2 = (d == 0) ? 2 : d - 1;   // (d+2)%3
    __syncthreads();  // prior reads of P / writes of T complete
    for (int k = h; k < K1; k += 2) {
      P[r * ldp + k] = XV[(d * 16 + r) * ldxv + k] * a0[r];
      P[r * ldp + K1 + k] = XV[(d1 * 16 + r) * ldxv + k] * a1[r * 3 + d2]
                          - XV[(d2 * 16 + r) * ldxv + k] * a1[r * 3 + d1];
    }
    __syncthreads();
    for (int c0 = 0; c0 < N1; c0 += 16) {
      v8f c;
      for (int j = 0; j < 8; ++j) {
        int row = j + 8 * h;
        c[j] = T[row * ldt + c0 + r] * a1[row * 3 + d];
      }
      c = wmma_acc(P,      ldp, K1, w.w10, N1, c0, c);
      c = wmma_acc(P + K1, ldp, K1, w.w11, N1, c0, c);
      for (int j = 0; j < 8; ++j) Vout[(d * 16 + (j + 8 * h)) * ldvo + c0 + r] = c[j];
    }
  }
  __syncthreads();
}

// _gate: S[:, :n0] -> silu, V *= sigmoid(S[:, n0:n0+n1])
__device__ void gate_wave(float* S, int lds_, float* V, int ldvo, int n0, int n1) {
  const int lane = threadIdx.x & 31;
  const int r = lane & 15;
  const int h = lane >> 4;
  __syncthreads();
  for (int c = h; c < n0; c += 2) {
    float x = S[r * lds_ + c];
    S[r * lds_ + c] = x / (1.f + __expf(-x));
  }
  for (int c = h; c < n1; c += 2) {
    float g = 1.f / (1.f + __expf(-S[r * lds_ + n0 + c]));
    for (int d = 0; d < 3; ++d) V[(d * 16 + r) * ldvo + c] *= g;
  }
  __syncthreads();
}

// ---------------------------------------------------------------------------
// Embed: per-node tiny tp (K0=2, K1=1) -> s[N][64], v[N][96] (d-major: d*32+i)
// node_attr a0 forced to 1.0 per reference.
// ---------------------------------------------------------------------------
__global__ void embed_kernel(const float* __restrict__ x, const float* __restrict__ nattr,
                             float* __restrict__ s, float* __restrict__ v,
                             TPW w, int N) {
  int n = blockIdx.x * blockDim.x + threadIdx.x;
  if (n >= N) return;
  float x0a = x[(size_t)n * 5 + 0], x0b = x[(size_t)n * 5 + 1];
  float vx = x[(size_t)n * 5 + 2], vy = x[(size_t)n * 5 + 3], vz = x[(size_t)n * 5 + 4];
  float a1x = nattr[(size_t)n * 4 + 1], a1y = nattr[(size_t)n * 4 + 2], a1z = nattr[(size_t)n * 4 + 3];
  float dt = vx * a1x + vy * a1y + vz * a1z;
  for (int o = 0; o < 64; ++o)
    s[(size_t)n * 64 + o] = x0a * w.w00[o] + x0b * w.w00[64 + o] + dt * w.w110[o] + w.b0[o];
  float cx = vy * a1z - vz * a1y;
  float cy = vz * a1x - vx * a1z;
  float cz = vx * a1y - vy * a1x;
  for (int o = 0; o < 32; ++o) {
    float t = x0a * w.w01[o] + x0b * w.w01[32 + o];
    v[(size_t)n * 96 + 0 * 32 + o] = t * a1x + vx * w.w10[o] + cx * w.w11[o];
    v[(size_t)n * 96 + 1 * 32 + o] = t * a1y + vy * w.w10[o] + cy * w.w11[o];
    v[(size_t)n * 96 + 2 * 32 + o] = t * a1z + vz * w.w10[o] + cz * w.w11[o];
  }
}

// ---------------------------------------------------------------------------
// Edge message kernel: gather -> m1 -> gate -> m2 -> gate -> atomic scatter.
// 16 edges per wave32 block.
// ---------------------------------------------------------------------------
__global__ __launch_bounds__(32) void edge_kernel(
    const float* __restrict__ s, const float* __restrict__ v,
    const int* __restrict__ ei, const float* __restrict__ ea,
    const float* __restrict__ amf,
    float* __restrict__ agg_s, float* __restrict__ agg_v,
    TPW m1, TPW m2, int E) {
  __shared__ float X0[16 * 132];       // [s_dst|s_src|amf] K0=129  -> reused as m2 S-out
  __shared__ float XV[3 * 16 * 68];    // [v_dst|v_src]     K1=64   -> reused as m2 V-out
  __shared__ float A0[16], A1[16 * 3];
  __shared__ float P[16 * 196];
  __shared__ float T[16 * 36];
  __shared__ float S1[16 * 100];
  __shared__ float V1[3 * 16 * 36];

  const int lane = threadIdx.x & 31;
  const int r = lane & 15;
  const int h = lane >> 4;
  int e = blockIdx.x * 16 + r;
  int ec = (e < E) ? e : 0;
  int src = ei[ec];
  int dst = ei[E + ec];

  if (h == 0) {
    A0[r] = ea[(size_t)ec * 4 + 0];
    A1[r * 3 + 0] = ea[(size_t)ec * 4 + 1];
    A1[r * 3 + 1] = ea[(size_t)ec * 4 + 2];
    A1[r * 3 + 2] = ea[(size_t)ec * 4 + 3];
    X0[r * 132 + 128] = amf[ec];
  }
  for (int k = h; k < 64; k += 2) {
    X0[r * 132 + k]      = s[(size_t)dst * 64 + k];
    X0[r * 132 + 64 + k] = s[(size_t)src * 64 + k];
  }
  for (int k = h; k < 32; k += 2)
    for (int d = 0; d < 3; ++d) {
      XV[(d * 16 + r) * 68 + k]      = v[(size_t)dst * 96 + d * 32 + k];
      XV[(d * 16 + r) * 68 + 32 + k] = v[(size_t)src * 96 + d * 32 + k];
    }
  __syncthreads();

  tp_wave(X0, 132, 129, XV, 68, 64, A0, A1, P, 196, T, 36, S1, 100, 96, V1, 36, 32, m1);
  gate_wave(S1, 100, V1, 36, 64, 32);
  // m2: inputs are S1/V1; original gather buffers are free -> use them as outputs
  tp_wave(S1, 100, 64, V1, 36, 32, A0, A1, P, 196, T, 36, X0, 132, 96, XV, 68, 32, m2);
  gate_wave(X0, 132, XV, 68, 64, 32);

  if (e < E) {
    for (int k = h; k < 64; k += 2)
      unsafeAtomicAdd(&agg_s[(size_t)dst * 64 + k], X0[r * 132 + k]);
    for (int k = h; k < 32; k += 2)
      for (int d = 0; d < 3; ++d)
        unsafeAtomicAdd(&agg_v[(size_t)dst * 96 + d * 32 + k], XV[(d * 16 + r) * 68 + k]);
  }
}

// ---------------------------------------------------------------------------
// Node update kernel: [s|agg_s],[v|agg_v] -> u1 -> gate -> u2 -> residual add.
// a0 = 1.0 (node_attr[:,0] forced to 1 in reference).
// ---------------------------------------------------------------------------
__global__ __launch_bounds__(32) void node_kernel(
    float* __restrict__ s, float* __restrict__ v,
    const float* __restrict__ agg_s, const float* __restrict__ agg_v,
    const float* __restrict__ nattr,
    TPW u1, TPW u2, int N) {
  __shared__ float X0[16 * 132];       // [s|agg_s] K0=128 -> reused as u2 S-out
  __shared__ float XV[3 * 16 * 68];    // [v|agg_v] K1=64  -> reused as u2 V-out
  __shared__ float A0[16], A1[16 * 3];
  __shared__ float P[16 * 196];
  __shared__ float T[16 * 36];
  __shared__ float S1[16 * 100];
  __shared__ float V1[3 * 16 * 36];

  const int lane = threadIdx.x & 31;
  const int r = lane & 15;
  const int h = lane >> 4;
  int n = blockIdx.x * 16 + r;
  int nc = (n < N) ? n : 0;

  if (h == 0) {
    A0[r] = 1.0f;
    A1[r * 3 + 0] = nattr[(size_t)nc * 4 + 1];
    A1[r * 3 + 1] = nattr[(size_t)nc * 4 + 2];
    A1[r * 3 + 2] = nattr[(size_t)nc * 4 + 3];
  }
  for (int k = h; k < 64; k += 2) {
    X0[r * 132 + k]      = s[(size_t)nc * 64 + k];
    X0[r * 132 + 64 + k] = agg_s[(size_t)nc * 64 + k];
  }
  for (int k = h; k < 32; k += 2)
    for (int d = 0; d < 3; ++d) {
      XV[(d * 16 + r) * 68 + k]      = v[(size_t)nc * 96 + d * 32 + k];
      XV[(d * 16 + r) * 68 + 32 + k] = agg_v[(size_t)nc * 96 + d * 32 + k];
    }
  __syncthreads();

  tp_wave(X0, 132, 128, XV, 68, 64, A0, A1, P, 196, T, 36, S1, 100, 96, V1, 36, 32, u1);
  gate_wave(S1, 100, V1, 36, 64, 32);
  tp_wave(S1, 100, 64, V1, 36, 32, A0, A1, P, 196, T, 36, X0, 132, 64, XV, 68, 32, u2);

  if (n < N) {
    for (int k = h; k < 64; k += 2)
      s[(size_t)n * 64 + k] += X0[r * 132 + k];
    for (int k = h; k < 32; k += 2)
      for (int d = 0; d < 3; ++d)
        v[(size_t)n * 96 + d * 32 + k] += XV[(d * 16 + r) * 68 + k];
  }
}

// ---------------------------------------------------------------------------
// Post-process: pp1 (WMMA tp + gate), pp2 (N0o=N1o=1, VALU), write out [N][4].
// ---------------------------------------------------------------------------
__global__ __launch_bounds__(32) void pp_kernel(
    const float* __restrict__ s, const float* __restrict__ v,
    const float* __restrict__ nattr,
    float* __restrict__ out, TPW pp1, TPW pp2, int N) {
  __shared__ float X0[16 * 68];
  __shared__ float XV[3 * 16 * 36];
  __shared__ float A0[16], A1[16 * 3];
  __shared__ float P[16 * 100];
  __shared__ float T[16 * 36];
  __shared__ float S1[16 * 100];
  __shared__ float V1[3 * 16 * 36];

  const int lane = threadIdx.x & 31;
  const int r = lane & 15;
  const int h = lane >> 4;
  int n = blockIdx.x * 16 + r;
  int nc = (n < N) ? n : 0;

  if (h == 0) {
    A0[r] = 1.0f;
    A1[r * 3 + 0] = nattr[(size_t)nc * 4 + 1];
    A1[r * 3 + 1] = nattr[(size_t)nc * 4 + 2];
    A1[r * 3 + 2] = nattr[(size_t)nc * 4 + 3];
  }
  for (int k = h; k < 64; k += 2) X0[r * 68 + k] = s[(size_t)nc * 64 + k];
  for (int k = h; k < 32; k += 2)
    for (int d = 0; d < 3; ++d)
      XV[(d * 16 + r) * 36 + k] = v[(size_t)nc * 96 + d * 32 + k];
  __syncthreads();

  tp_wave(X0, 68, 64, XV, 36, 32, A0, A1, P, 100, T, 36, S1, 100, 96, V1, 36, 32, pp1);
  gate_wave(S1, 100, V1, 36, 64, 32);

  if (h == 0 && n < N) {
    float a1x = A1[r * 3 + 0], a1y = A1[r * 3 + 1], a1z = A1[r * 3 + 2];
    float acc0 = pp2.b0[0], t = 0.f;
    for (int k = 0; k < 64; ++k) {
      float f = S1[r * 100 + k];
      acc0 += f * pp2.w00[k];
      t    += f * pp2.w01[k];
    }
    float ox = t * a1x, oy = t * a1y, oz = t * a1z;
    for (int k = 0; k < 32; ++k) {
      float vx = V1[(0 * 16 + r) * 36 + k];
      float vy = V1[(1 * 16 + r) * 36 + k];
      float vz = V1[(2 * 16 + r) * 36 + k];
      acc0 += (vx * a1x + vy * a1y + vz * a1z) * pp2.w110[k];
      ox += vx * pp2.w10[k] + (vy * a1z - vz * a1y) * pp2.w11[k];
      oy += vy * pp2.w10[k] + (vz * a1x - vx * a1z) * pp2.w11[k];
      oz += vz * pp2.w10[k] + (vx * a1y - vy * a1x) * pp2.w11[k];
    }
    out[(size_t)n * 4 + 0] = acc0;
    out[(size_t)n * 4 + 1] = ox;
    out[(size_t)n * 4 + 2] = oy;
    out[(size_t)n * 4 + 3] = oz;
  }
}

// ---------------------------------------------------------------------------
// Host side
// ---------------------------------------------------------------------------
static void tp_leaf_sizes(int n0i, int n1i, int n0o, int n1o, int* sz) {
  // insertion order: w00, w110, b0, w01, w10, w11
  sz[0] = n0i * n0o; sz[1] = n1i * n0o; sz[2] = n0o;
  sz[3] = n0i * n1o; sz[4] = n1i * n1o; sz[5] = n1i * n1o;
}

static TPW mk_tpw(const float* const* L, int i) {
  TPW w;
  w.w00 = L[i + 0]; w.w110 = L[i + 1]; w.b0 = L[i + 2];
  w.w01 = L[i + 3]; w.w10  = L[i + 4]; w.w11 = L[i + 5];
  return w;
}

extern "C" void kernel_launch(void* const* d_in, const int* in_sizes, int n_in,
                              void* d_out, int out_size, void* d_ws, size_t ws_size,
                              hipStream_t stream) {
  const float* x     = (const float*)d_in[0];
  const int*   ei    = (const int*)d_in[1];
  const float* ea    = (const float*)d_in[2];
  const float* nattr = (const float*)d_in[3];
  const float* amf   = (const float*)d_in[5];
  const int N = in_sizes[0] / 5;   // x: [N, 5]
  const int E = in_sizes[2] / 4;   // edge_attr: [E, 4]

  // tp dims in params insertion order: embed, (m1,m2,u1,u2) x 2 layers, pp1, pp2
  static const int dims[11][4] = {
      {2, 1, 64, 32},
      {129, 64, 96, 32}, {64, 32, 96, 32}, {128, 64, 96, 32}, {64, 32, 64, 32},
      {129, 64, 96, 32}, {64, 32, 96, 32}, {128, 64, 96, 32}, {64, 32, 64, 32},
      {64, 32, 96, 32},
      {64, 32, 1, 1}};

  const float* L[66];
  if (n_in >= 72) {
    for (int i = 0; i < 66; ++i) L[i] = (const float*)d_in[6 + i];
  } else {
    // single concatenated params buffer
    const float* base = (const float*)d_in[6];
    size_t off = 0;
    int li = 0;
    for (int t = 0; t < 11; ++t) {
      int sz[6];
      tp_leaf_sizes(dims[t][0], dims[t][1], dims[t][2], dims[t][3], sz);
      for (int j = 0; j < 6; ++j) { L[li++] = base + off; off += (size_t)sz[j]; }
    }
  }

  TPW w_embed = mk_tpw(L, 0);
  TPW w_m1[2] = {mk_tpw(L, 6),  mk_tpw(L, 30)};
  TPW w_m2[2] = {mk_tpw(L, 12), mk_tpw(L, 36)};
  TPW w_u1[2] = {mk_tpw(L, 18), mk_tpw(L, 42)};
  TPW w_u2[2] = {mk_tpw(L, 24), mk_tpw(L, 48)};
  TPW w_pp1 = mk_tpw(L, 54);
  TPW w_pp2 = mk_tpw(L, 60);

  // workspace: s[N][64], v[N][96], agg_s[N][64], agg_v[N][96]
  float* s_buf = (float*)d_ws;
  float* v_buf = s_buf + (size_t)N * 64;
  float* ag_s  = v_buf + (size_t)N * 96;
  float* ag_v  = ag_s + (size_t)N * 64;

  embed_kernel<<<(N + 255) / 256, 256, 0, stream>>>(x, nattr, s_buf, v_buf, w_embed, N);

  const int eBlocks = (E + 15) / 16;
  const int nBlocks = (N + 15) / 16;
  for (int l = 0; l < 2; ++l) {
    hipMemsetAsync(ag_s, 0, (size_t)N * 64 * sizeof(float), stream);
    hipMemsetAsync(ag_v, 0, (size_t)N * 96 * sizeof(float), stream);
    edge_kernel<<<eBlocks, 32, 0, stream>>>(s_buf, v_buf, ei, ea, amf, ag_s, ag_v,
                                            w_m1[l], w_m2[l], E);
    node_kernel<<<nBlocks, 32, 0, stream>>>(s_buf, v_buf, ag_s, ag_v, nattr,
                                            w_u1[l], w_u2[l], N);
  }
  pp_kernel<<<nBlocks, 32, 0, stream>>>(s_buf, v_buf, nattr, (float*)d_out, w_pp1, w_pp2, N);

  (void)in_sizes; (void)n_in; (void)out_size; (void)ws_size;
}